// EGACECLoss_51539607552337
// MI455X (gfx1250) — compile-verified
//
#include <hip/hip_runtime.h>

// ---------------------------------------------------------------------------
// EGACECLoss on MI455X (gfx1250):
//   out = sum(Z*Z) - || Z^T P ||_F^2,  Z:[N,256] f32, P:[N,20] f32
//
// Memory-bound: ~221 MB mandatory reads -> ~9.5us at 23.3 TB/s. Single pass
// over Z; Z^T P uses native fp32 WMMA V_WMMA_F32_16X16X4_F32 so the matrix
// pipes do the outer-product accumulation at full fp32 precision.
// K=20 padded to 32; padding columns are fed exact zeros via a per-lane
// 0/1 mask multiply (branchless - keeps EXEC untouched in the hot loop).
// ---------------------------------------------------------------------------

#define D_DIM   256
#define K_DIM   20
#define KPAD    32
#define DTILES  16            // 256 / 16
#define NCHUNKS 250           // 200000 / 250 = 800 rows/chunk (divisible by 16)
#define WAVES_PER_BLOCK 8     // 256 threads = 8 wave32 on CDNA5

typedef float v2f __attribute__((ext_vector_type(2)));
typedef float v8f __attribute__((ext_vector_type(8)));

// ws layout (floats): [0] = term1 accumulator, [16 .. 16+256*32) = ztp tiles
#define WS_FLOATS (16 + D_DIM * KPAD)

__global__ void egace_zero_ws(float* ws, int n) {
    int i = blockIdx.x * blockDim.x + threadIdx.x;
    if (i < n) ws[i] = 0.0f;
}

__global__ __launch_bounds__(256)
void egace_ztp_wmma(const float* __restrict__ Z,
                    const float* __restrict__ P,
                    float* __restrict__ ws, int N) {
    const int lane   = threadIdx.x & 31;
    const int laneM  = lane & 15;       // M (A) / N-col (B,C,D) index
    const int laneHi = lane >> 4;       // selects K=2,3 half per ISA fp32 layout
    const int wid    = (blockIdx.x * blockDim.x + threadIdx.x) >> 5;
    const int dtile  = wid & (DTILES - 1);
    const int chunk  = wid / DTILES;
    const int d0     = dtile * 16;
    const int CH     = N / NCHUNKS;     // 800
    const int nbeg   = chunk * CH;
    const int nend   = nbeg + CH;

    // Per-lane constant offsets for the WMMA fp32 A/B register layouts:
    //   A VGPR v, lane L: element Z[n0 + 2*(L>=16) + v][d0 + L%16]
    //   B VGPR v, lane L: element P[n0 + 2*(L>=16) + v][k0 + L%16]
    const int zoff0 = (2 * laneHi) * D_DIM + d0 + laneM;
    const int zoff1 = zoff0 + D_DIM;
    const int poff0 = (2 * laneHi) * K_DIM + laneM;
    const int poff1 = poff0 + K_DIM;

    // K-tile1 covers k=16..31; only k=16..19 exist. Branchless handling:
    // always load a VALID address (clamped back into the row for lanes whose
    // column would be OOB) and multiply by a 0/1 mask. No EXEC manipulation
    // in the hot loop, loads stay clause-batched.
    const bool k1valid  = laneM < (K_DIM - 16);           // laneM < 4
    const int  poff1c_0 = (2 * laneHi) * K_DIM + (k1valid ? laneM + 16 : laneM);
    const int  poff1c_1 = poff1c_0 + K_DIM;
    const float m1 = k1valid ? 1.0f : 0.0f;

    v8f acc0 = {0.f, 0.f, 0.f, 0.f, 0.f, 0.f, 0.f, 0.f};  // k = 0..15
    v8f acc1 = {0.f, 0.f, 0.f, 0.f, 0.f, 0.f, 0.f, 0.f};  // k = 16..31 (cols >=20 stay 0)
    float t1 = 0.0f;

    for (int n0 = nbeg; n0 < nend; n0 += 16) {
        // speculative prefetch of next Z block (global_prefetch_b8)
        __builtin_prefetch(&Z[(n0 + 16) * D_DIM + zoff0], 0, 3);
#pragma unroll
        for (int s = 0; s < 4; ++s) {
            const int nb = n0 + s * 4;
            const int zb = nb * D_DIM;
            const int pb = nb * K_DIM;

            v2f a;
            a.x = Z[zb + zoff0];
            a.y = Z[zb + zoff1];

            v2f b0;
            b0.x = P[pb + poff0];
            b0.y = P[pb + poff1];

            v2f b1;
            b1.x = P[pb + poff1c_0] * m1;
            b1.y = P[pb + poff1c_1] * m1;

            // fuse term1 = sum(Z*Z): every Z element is loaded exactly once grid-wide
            t1 += a.x * a.x + a.y * a.y;

            // D = A(16x4) * B(4x16) + C, fp32 throughout
            acc0 = __builtin_amdgcn_wmma_f32_16x16x4_f32(
                false, a, false, b0, (short)0, acc0, false, false);
            acc1 = __builtin_amdgcn_wmma_f32_16x16x4_f32(
                false, a, false, b1, (short)0, acc1, false, false);
        }
    }

    // wave32 reduction of term1, one atomic per wave
    for (int off = 16; off > 0; off >>= 1)
        t1 += __shfl_xor(t1, off, 32);
    if (lane == 0) atomicAdd(&ws[0], t1);

    // scatter partial ztp tiles: C/D layout -> row M = v + 8*(lane>=16), col = lane%16
    float* ztp = ws + 16;
#pragma unroll
    for (int v = 0; v < 8; ++v) {
        const int d = d0 + v + 8 * laneHi;
        atomicAdd(&ztp[d * KPAD + laneM],      acc0[v]);
        atomicAdd(&ztp[d * KPAD + 16 + laneM], acc1[v]);
    }
}

__global__ __launch_bounds__(256)
void egace_finalize(const float* __restrict__ ws, float* __restrict__ out) {
    __shared__ float red[8];
    const float* ztp = ws + 16;
    float s = 0.0f;
#pragma unroll 4
    for (int j = 0; j < 32; ++j) {
        float v = ztp[threadIdx.x + 256 * j];   // 8192 = 256*32 elements, coalesced
        s += v * v;
    }
    for (int off = 16; off > 0; off >>= 1)
        s += __shfl_xor(s, off, 32);
    const int w = threadIdx.x >> 5;
    if ((threadIdx.x & 31) == 0) red[w] = s;
    __syncthreads();
    if (threadIdx.x == 0) {
        float t2 = 0.0f;
#pragma unroll
        for (int i = 0; i < 8; ++i) t2 += red[i];
        out[0] = ws[0] - t2;
    }
}

extern "C" void kernel_launch(void* const* d_in, const int* in_sizes, int n_in,
                              void* d_out, int out_size, void* d_ws, size_t ws_size,
                              hipStream_t stream) {
    const float* Z = (const float*)d_in[0];   // embeddings [N, 256]
    const float* P = (const float*)d_in[1];   // indicator  [N, 20]
    float* out = (float*)d_out;
    float* ws  = (float*)d_ws;
    const int N = in_sizes[0] / D_DIM;        // 200000

    // zero workspace every call (harness poisons ws; replays must be deterministic)
    egace_zero_ws<<<(WS_FLOATS + 255) / 256, 256, 0, stream>>>(ws, WS_FLOATS);

    const int totalWaves = DTILES * NCHUNKS;              // 4000 waves
    const int blocks = totalWaves / WAVES_PER_BLOCK;      // 500 blocks x 256 thr
    egace_ztp_wmma<<<blocks, 256, 0, stream>>>(Z, P, ws, N);

    egace_finalize<<<1, 256, 0, stream>>>(ws, out);
}